// BaseModel_74526272520550
// MI455X (gfx1250) — compile-verified
//
#include <hip/hip_runtime.h>

// ---------------------------------------------------------------------------
// CDNA5 WMMA f32 16x16x4 fragment types:
//   A (16x4 f32): 2 VGPRs/lane -> v2f   B (4x16 f32): 2 VGPRs/lane -> v2f
//   C/D (16x16 f32): 8 VGPRs/lane -> v8f
// ---------------------------------------------------------------------------
typedef __attribute__((ext_vector_type(2))) float v2f;
typedef __attribute__((ext_vector_type(4))) float v4f;
typedef __attribute__((ext_vector_type(8))) float v8f;

#define FEAT 100          // (1+1+3+5)^2 feature dim
#define WROWS 112         // 100 output cols padded to 7*16
#define W_ELEMS (WROWS * FEAT)
#define TILE_STRIDE (WROWS * 16)   // floats per 16-edge tile in `pre`

// Angular-momentum layout tables (COMMON_AM = [0,0,1,2])
__constant__ int c_D[4]  = {1, 1, 3, 5};   // 2l+1 per position
__constant__ int c_PS[4] = {0, 1, 2, 5};   // prefix sums of c_D
// row/col -> (am position, index within 2l+1) per atom type
__constant__ int c_pmap0[4]  = {0, 2, 2, 2};
__constant__ int c_imap0[4]  = {0, 0, 1, 2};
__constant__ int c_pmap1[10] = {0, 1, 2, 2, 2, 3, 3, 3, 3, 3};
__constant__ int c_imap1[10] = {0, 0, 0, 1, 2, 0, 1, 2, 3, 4};

// ---------------------------------------------------------------------------
// Kernel 1: expand CG tensors into dense block-diagonal weight W[p][q]
// W[bs + i*c + j][bs + k] = cg_{l1,l2}[i][j][k], bs = 10*PS[p1] + D[p1]*PS[p2]
// ---------------------------------------------------------------------------
__global__ void k_wbuild(const float* cg00, const float* cg01, const float* cg02,
                         const float* cg10, const float* cg11, const float* cg12,
                         const float* cg20, const float* cg21, const float* cg22,
                         float* __restrict__ Wg) {
  for (int idx = threadIdx.x; idx < W_ELEMS; idx += blockDim.x) Wg[idx] = 0.0f;
  __syncthreads();
  const float* cg[9] = {cg00, cg01, cg02, cg10, cg11, cg12, cg20, cg21, cg22};
  const int AM[4] = {0, 0, 1, 2};
  const int Dl[4] = {1, 1, 3, 5};
  const int PSl[4] = {0, 1, 2, 5};
  int t = threadIdx.x;
  if (t < 16) {
    int p1 = t >> 2, p2 = t & 3;
    int l1 = AM[p1], l2 = AM[p2];
    int r = 2 * l1 + 1, c = 2 * l2 + 1, sz = r * c;
    int bs = 10 * PSl[p1] + Dl[p1] * PSl[p2];
    const float* g = cg[l1 * 3 + l2];
    for (int i = 0; i < r; ++i)
      for (int j = 0; j < c; ++j)
        for (int k = 0; k < sz; ++k)
          Wg[(bs + i * c + j) * FEAT + (bs + k)] = g[(i * c + j) * sz + k];
  }
}

// ---------------------------------------------------------------------------
// Kernel 2: g2b ranks, written as f32 into the tail of d_out
// ---------------------------------------------------------------------------
__global__ void k_g2b(const int* __restrict__ s00, const int* __restrict__ s01,
                      const int* __restrict__ s10, const int* __restrict__ s11,
                      int n00, int n01, int n10, int n11, float* __restrict__ gout) {
  int tid = blockIdx.x * blockDim.x + threadIdx.x;
  int T0 = n00, T1 = T0 + n01, T2 = T1 + n10, T3 = T2 + n11;
  if (tid >= T3) return;
  const int* s;
  int i;
  if (tid < T0)      { s = s00; i = tid; }
  else if (tid < T1) { s = s01; i = tid - T0; }
  else if (tid < T2) { s = s10; i = tid - T1; }
  else               { s = s11; i = tid - T2; }
  gout[s[i]] = (float)i;
}

// ---------------------------------------------------------------------------
// WMMA inner loop (templated on whether any lane in the wave is a self edge,
// so the common no-self path carries no exec-mask juggling at all).
// ---------------------------------------------------------------------------
template <bool SELF>
__device__ __forceinline__ void mm_tile(const float* __restrict__ rowFE,
                                        const float* __restrict__ rowFN,
                                        bool is_self,
                                        const float* Wl,  // LDS, p-major
                                        int m, int kh, v8f acc[7]) {
  for (int kk = 0; kk < 25; ++kk) {
    int koff = kk * 4 + kh * 2;                        // even -> 8B aligned
    v2f a = *(const v2f*)(rowFE + koff);               // A frag
    if (SELF) {
      if (is_self) a += *(const v2f*)(rowFN + koff);   // divergence pre-WMMA only
    }
#pragma unroll
    for (int n = 0; n < 7; ++n) {
      // B frag: B[kr][nc] = W[16n+nc][k0+kr]; lane holds kr = {2kh, 2kh+1}
      v2f b = *(const v2f*)(&Wl[(n * 16 + m) * FEAT + koff]);
      acc[n] = __builtin_amdgcn_wmma_f32_16x16x4_f32(
          false, a, false, b, (short)0, acc[n], false, false);
    }
  }
}

// ---------------------------------------------------------------------------
// Kernel 3: WMMA GEMM  pre = FE' @ W^T, tile-major output layout:
//   pre[tile][p (0..111)][row (0..15)]  -> 2x global_store_b128 per N-tile,
//   fully coalesced across the wave, no store guards needed.
// ---------------------------------------------------------------------------
__global__ __launch_bounds__(256) void k_blocks(
    const float* __restrict__ fe, const float* __restrict__ fn,
    const float* __restrict__ S, const int* __restrict__ n1,
    const int* __restrict__ n2, const float* __restrict__ Wg,
    float* __restrict__ pre, int E) {
  __shared__ float Wl[W_ELEMS];  // 44.8 KB, p-major: Wl[p*100 + q]
  for (int idx = threadIdx.x; idx < W_ELEMS; idx += blockDim.x) Wl[idx] = Wg[idx];
  __syncthreads();

  int wave = threadIdx.x >> 5;   // wave32
  int lane = threadIdx.x & 31;
  int tile = blockIdx.x * 8 + wave;
  long base = (long)tile * 16;
  if (base >= E) return;         // wave-uniform: EXEC stays all-ones for WMMA

  int m  = lane & 15;            // edge-in-tile (A row) / B column
  int kh = lane >> 4;            // K-half (A layout: lanes 16-31 hold K=2,3)
  long e = base + m; if (e > E - 1) e = E - 1;   // clamp keeps EXEC full

  const float* rowFE = fe + e * FEAT;
  int a1 = n1[e], a2 = n2[e];
  float sx = S[e * 3 + 0], sy = S[e * 3 + 1], sz = S[e * 3 + 2];
  bool is_self = (a1 == a2) && (sx * sx + sy * sy + sz * sz < 1e-12f);
  const float* rowFN = fn + (long)a1 * FEAT;

  v8f acc[7];
#pragma unroll
  for (int n = 0; n < 7; ++n) acc[n] = (v8f){0, 0, 0, 0, 0, 0, 0, 0};

  if (__any(is_self))            // wave-uniform branch
    mm_tile<true>(rowFE, rowFN, is_self, Wl, m, kh, acc);
  else
    mm_tile<false>(rowFE, rowFN, is_self, Wl, m, kh, acc);

  // C/D layout: VGPR r -> (M = r + 8*kh, N = m); store octet contiguously.
  long tbase = (long)tile * TILE_STRIDE;
#pragma unroll
  for (int n = 0; n < 7; ++n) {
    long idx = tbase + (long)(n * 16 + m) * 16 + kh * 8;
    v4f lo = {acc[n][0], acc[n][1], acc[n][2], acc[n][3]};
    v4f hi = {acc[n][4], acc[n][5], acc[n][6], acc[n][7]};
    *(v4f*)(pre + idx)     = lo;
    *(v4f*)(pre + idx + 4) = hi;
  }
}

// ---------------------------------------------------------------------------
// Kernel 4: assemble + hermitian-symmetrize:
// hams_g[i][r][c] = 0.5 * (pre[e][p(g,r,c)] + pre[inv(e)][p(g',c,r)])
// pre is tile-major: pre[(e>>4)*1792 + p*16 + (e&15)]
// ---------------------------------------------------------------------------
__global__ void k_assemble(const float* __restrict__ pre, const int* __restrict__ inv,
                           const int* __restrict__ s00, const int* __restrict__ s01,
                           const int* __restrict__ s10, const int* __restrict__ s11,
                           int n00, int n01, int n10, int n11,
                           float* __restrict__ out) {
  long long tid = (long long)blockIdx.x * blockDim.x + threadIdx.x;
  long long T0 = (long long)n00 * 16;
  long long T1 = T0 + (long long)n01 * 40;
  long long T2 = T1 + (long long)n10 * 40;
  long long T3 = T2 + (long long)n11 * 100;
  if (tid >= T3) return;

  int t1, t2, C;
  const int* sel;
  long long local, obase;
  if (tid < T0)      { t1 = 0; t2 = 0; C = 4;  sel = s00; local = tid;      obase = 0;  }
  else if (tid < T1) { t1 = 0; t2 = 1; C = 10; sel = s01; local = tid - T0; obase = T0; }
  else if (tid < T2) { t1 = 1; t2 = 0; C = 4;  sel = s10; local = tid - T1; obase = T1; }
  else               { t1 = 1; t2 = 1; C = 10; sel = s11; local = tid - T2; obase = T2; }
  int RC = (t1 ? 10 : 4) * C;
  long long i = local / RC;
  int rc = (int)(local % RC);
  int r = rc / C, c = rc % C;

  long e  = sel[i];
  long e2 = inv[e];

  int p1 = (t1 ? c_pmap1 : c_pmap0)[r];
  int ii = (t1 ? c_imap1 : c_imap0)[r];
  int p2 = (t2 ? c_pmap1 : c_pmap0)[c];
  int jj = (t2 ? c_imap1 : c_imap0)[c];

  int p  = 10 * c_PS[p1] + c_D[p1] * c_PS[p2] + ii * c_D[p2] + jj;
  int pT = 10 * c_PS[p2] + c_D[p2] * c_PS[p1] + jj * c_D[p1] + ii;

  long pe  = (e  >> 4) * (long)TILE_STRIDE + p  * 16 + (e  & 15);
  long pe2 = (e2 >> 4) * (long)TILE_STRIDE + pT * 16 + (e2 & 15);
  out[obase + local] = 0.5f * (pre[pe] + pre[pe2]);
}

// ---------------------------------------------------------------------------
extern "C" void kernel_launch(void* const* d_in, const int* in_sizes, int n_in,
                              void* d_out, int out_size, void* d_ws, size_t ws_size,
                              hipStream_t stream) {
  const float* feature_node = (const float*)d_in[0];
  const float* feature_edge = (const float*)d_in[1];
  const float* S_hop        = (const float*)d_in[2];
  const int*   edge_index   = (const int*)d_in[3];
  // d_in[4] = AtomType (unused: selections are precomputed)
  const int*   edge_inv     = (const int*)d_in[5];
  const float* cg[9];
  for (int i = 0; i < 9; ++i) cg[i] = (const float*)d_in[6 + i];
  const int* s00 = (const int*)d_in[15];
  const int* s01 = (const int*)d_in[16];
  const int* s10 = (const int*)d_in[17];
  const int* s11 = (const int*)d_in[18];
  int n00 = in_sizes[15], n01 = in_sizes[16], n10 = in_sizes[17], n11 = in_sizes[18];

  int E = in_sizes[5];               // edge_inverse has E entries
  const int* n1 = edge_index;
  const int* n2 = edge_index + E;

  // Workspace: [W (112x100 f32)] [pad to 256B] [pre (tiles x 112 x 16 f32)]
  float* Wg  = (float*)d_ws;
  size_t pre_off = ((size_t)W_ELEMS * sizeof(float) + 255) & ~(size_t)255;
  float* pre = (float*)((char*)d_ws + pre_off);

  long long T3 = (long long)n00 * 16 + (long long)n01 * 40 +
                 (long long)n10 * 40 + (long long)n11 * 100;
  float* g2b_out = (float*)d_out + T3;

  // 1) Build dense weight matrix from CG tensors
  k_wbuild<<<1, 256, 0, stream>>>(cg[0], cg[1], cg[2], cg[3], cg[4],
                                  cg[5], cg[6], cg[7], cg[8], Wg);

  // 2) g2b ranks -> tail of d_out
  {
    int total = n00 + n01 + n10 + n11;
    int blocks = (total + 255) / 256;
    k_g2b<<<blocks, 256, 0, stream>>>(s00, s01, s10, s11, n00, n01, n10, n11, g2b_out);
  }

  // 3) WMMA block contraction -> pre (tile-major)
  {
    int tiles = (E + 15) / 16;
    int wgs = (tiles + 7) / 8;
    k_blocks<<<wgs, 256, 0, stream>>>(feature_edge, feature_node, S_hop,
                                      n1, n2, Wg, pre, E);
  }

  // 4) Assemble hamiltonian blocks with hermitian symmetrization -> d_out
  {
    long long blocks = (T3 + 255) / 256;
    k_assemble<<<(unsigned)blocks, 256, 0, stream>>>(pre, edge_inv,
                                                     s00, s01, s10, s11,
                                                     n00, n01, n10, n11,
                                                     (float*)d_out);
  }
}